// GWCBuilder_22488448761896
// MI455X (gfx1250) — compile-verified
//
#include <hip/hip_runtime.h>

typedef float v2f __attribute__((ext_vector_type(2)));
typedef float v8f __attribute__((ext_vector_type(8)));

#define B_N 4
#define C_N 320
#define H_N 80
#define W_N 160
#define G_N 40
#define CPG 8
#define D_N 48
#define TI_N (W_N / 16)      // 10 w-tiles per row
#define HW   (H_N * W_N)     // 12800
#define WPB  8               // waves per block
#define LDS_STRIDE (D_N * 16 + 32)  // 768 payload + 32 trash floats

// One wave handles one (b, g, h, w-tile). Banded-matmul formulation:
//   A[w, c] = L[c, w]  (16x4 per WMMA), B[c, w'] = R[c, w']  (4x16)
//   (A*B)[w, w'] = sum_c L[c,w] R[c,w'] = 8 * cost[d = w - w', w]
// Two V_WMMA_F32_16X16X4_F32 per B-tile accumulate the cpg=8 reduction.
__global__ __launch_bounds__(256) void gwc_cost_volume_kernel(
    const float* __restrict__ L, const float* __restrict__ R,
    float* __restrict__ out)
{
    __shared__ __align__(16) float lds[WPB * LDS_STRIDE];

    const int lane = threadIdx.x & 31;
    const int wib  = threadIdx.x >> 5;
    const int wave = blockIdx.x * WPB + wib;

    // wave -> (b, g, h, ti)
    const int ti = wave % TI_N;
    int t = wave / TI_N;
    const int h = t % H_N;  t /= H_N;
    const int g = t % G_N;
    const int b = t / G_N;

    float* tile = &lds[wib * LDS_STRIDE];

    // Pre-zero the (d, w) staging tile: supplies zeros for w < d planes.
#pragma unroll
    for (int i = 0; i < (D_N * 16) / 32; ++i)
        tile[i * 32 + lane] = 0.0f;

    // WMMA 16x16x4 f32 operand layout (ISA 7.12.2):
    //  A 16x4: lanes 0-15 -> row m=lane, VGPR0=K0,VGPR1=K1; lanes 16-31 -> K2,K3
    //  B 4x16: lanes carry col n=lane%16; VGPR0 holds row K=2*half, VGPR1 K=2*half+1
    const int half = lane >> 4;       // 0/1 selects K pair
    const int col  = lane & 15;       // m for A, n for B
    const int k0   = half * 2;        // base channel of this lane's K pair

    const int base = ((b * C_N + g * CPG) * H_N + h) * W_N;
    const int wL   = ti * 16 + col;

    v2f a0, a1;  // channels {k0,k0+1} and {k0+4,k0+5}
    a0.x = L[base + (k0 + 0) * HW + wL];
    a0.y = L[base + (k0 + 1) * HW + wL];
    a1.x = L[base + (k0 + 4) * HW + wL];
    a1.y = L[base + (k0 + 5) * HW + wL];

    // B-tiles J = ti-jj, jj = 0..3, intersect the d in [0,48) band.
    for (int jj = 0; jj < 4; ++jj) {
        const int J = ti - jj;
        if (J < 0) break;            // w' < 0 -> zero padding (pre-zeroed LDS)
        const int wR = J * 16 + col;

        v2f b0, b1;
        b0.x = R[base + (k0 + 0) * HW + wR];
        b0.y = R[base + (k0 + 1) * HW + wR];
        b1.x = R[base + (k0 + 4) * HW + wR];
        b1.y = R[base + (k0 + 5) * HW + wR];

        v8f acc = {};
        acc = __builtin_amdgcn_wmma_f32_16x16x4_f32(
            false, a0, false, b0, (short)0, acc, false, false);
        acc = __builtin_amdgcn_wmma_f32_16x16x4_f32(
            false, a1, false, b1, (short)0, acc, false, false);

        // Scatter D-tile -> (d, w) LDS layout. VGPR r holds rows m=r (lanes
        // 0-15) and m=r+8 (lanes 16-31), column n=col. d = 16*jj + m - n.
        // Branch-free masking via trash slot keeps EXEC all-ones for WMMA.
        const int dBase = 16 * jj;
#pragma unroll
        for (int r = 0; r < 8; ++r) {
            const int m  = r + half * 8;
            const int d  = dBase + m - col;
            const bool ok = (d >= 0) & (d < D_N);
            const int idx = ok ? (d * 16 + m) : (D_N * 16 + lane);
            tile[idx] = acc[r] * 0.125f;   // mean over cpg=8
        }
    }

    // Per-wave private LDS region; just drain DS pipe before re-reading.
    asm volatile("s_wait_dscnt 0" ::: "memory");

    // Coalesced writeout: 4 lanes cover one 16-float d-row (64B cachelines),
    // 8 d-rows per b128 store, 6 iterations cover d = 0..47.
    const int dq = lane >> 2;   // 0..7  d-row within iteration
    const int q  = lane & 3;    // 0..3  16B quad within row
    const int outBase = (b * G_N + g) * D_N * HW + h * W_N + ti * 16 + q * 4;
#pragma unroll
    for (int it = 0; it < 6; ++it) {
        const int d = it * 8 + dq;
        const float4 v = *(const float4*)&tile[d * 16 + q * 4];
        *(float4*)&out[outBase + d * HW] = v;
    }
}

extern "C" void kernel_launch(void* const* d_in, const int* in_sizes, int n_in,
                              void* d_out, int out_size, void* d_ws, size_t ws_size,
                              hipStream_t stream) {
    const float* L = (const float*)d_in[0];
    const float* R = (const float*)d_in[1];
    float* out = (float*)d_out;

    const int totalWaves = B_N * G_N * H_N * TI_N;   // 128000
    const int blocks = totalWaves / WPB;             // 16000
    gwc_cost_volume_kernel<<<blocks, WPB * 32, 0, stream>>>(L, R, out);
}